// DeformableAttention_9174050144402
// MI455X (gfx1250) — compile-verified
//
#include <hip/hip_runtime.h>
#include <hip/hip_bf16.h>

typedef __attribute__((ext_vector_type(2))) float v2f;
typedef __attribute__((ext_vector_type(8))) float v8f;

#define N_HEADS 8
#define N_POINTS 4
#define BSZ 8
#define QLEN 2048
#define DMODEL 256
#define DHEAD 32
#define KMAX 256

// ---------------------------------------------------------------------------
// FP32 WMMA GEMM:  C[M,N] = A[M,K] @ B[K,N] + bias[N]
// Block = 8 waves, all on ONE column group of width 16*NT; each wave owns a
// (16*MT)-row slice -> block tile is (8*16*MT) x (16*NT).
// The whole B panel [K x 16*NT] is staged once into LDS in *fragment order*:
//   lds[((k>>2)*COLSPAN + c)*4 + ((k>>1)&1)*2 + (k&1)] = B[k][col0+c]
// so a wave's per-K-step B fragment (V_WMMA_F32_16X16X4_F32 layout: lanes
// 0-15 hold K=kk,kk+1, lanes 16-31 hold K=kk+2,kk+3) is one ds_load_b64,
// bank-conflict-free (half=0 -> banks {4c,4c+1}, half=1 -> {4c+2,4c+3}).
// Steady-state inner loop: MT global_load_b64 (A) + NT ds_load_b64 (B)
// per MT*NT WMMAs.
// Requires: M%(16*MT)==0, N%(16*NT)==0, K%4==0, K<=KMAX, blockDim.x==256.
// ---------------------------------------------------------------------------
template <int MT, int NT>
__global__ __launch_bounds__(256) void gemm_wmma_f32(
    const float* __restrict__ A, const float* __restrict__ Bm,
    const float* __restrict__ bias, float* __restrict__ C,
    int M, int N, int K) {
  constexpr int COLSPAN = 16 * NT;
  __shared__ float ldsB[KMAX * COLSPAN];

  const int lane  = threadIdx.x & 31;
  const int wvIdx = threadIdx.x >> 5;

  const int colGroups = N / COLSPAN;
  const int colGroup  = blockIdx.x % colGroups;
  const int rowBlk    = blockIdx.x / colGroups;
  const int col0      = colGroup * COLSPAN;
  const int row0      = rowBlk * (8 * 16 * MT) + wvIdx * (16 * MT);
  const int m         = lane & 15;
  const int half      = lane >> 4;

  // ---- cooperative, coalesced fill of the B panel into LDS (fragment order)
  for (int idx = threadIdx.x; idx < K * COLSPAN; idx += 256) {
    const int k = idx / COLSPAN;
    const int c = idx - k * COLSPAN;
    const int ld = ((k >> 2) * COLSPAN + c) * 4 + ((k >> 1) & 1) * 2 + (k & 1);
    ldsB[ld] = Bm[(size_t)k * N + col0 + c];
  }
  __syncthreads();

  if (row0 >= M) return;  // idle tail waves (after the fill + barrier)

  v8f acc[MT][NT] = {};

  const float* __restrict__ Arow[MT];
#pragma unroll
  for (int mt = 0; mt < MT; ++mt)
    Arow[mt] = A + (size_t)(row0 + mt * 16 + m) * K;

  for (int kk = 0; kk < K; kk += 4) {
    const int kb = kk + half * 2;

    v2f a[MT];
#pragma unroll
    for (int mt = 0; mt < MT; ++mt)
      a[mt] = *(const v2f*)(Arow[mt] + kb);          // global_load_b64

    const float* __restrict__ bbase =
        ldsB + ((kk >> 2) * COLSPAN + m) * 4 + half * 2;
#pragma unroll
    for (int t = 0; t < NT; ++t) {
      const v2f b = *(const v2f*)(bbase + t * 64);   // ds_load_b64
#pragma unroll
      for (int mt = 0; mt < MT; ++mt) {
        acc[mt][t] = __builtin_amdgcn_wmma_f32_16x16x4_f32(
            /*neg_a=*/false, a[mt], /*neg_b=*/false, b,
            /*c_mod=*/(short)0, acc[mt][t],
            /*reuse_a=*/false, /*reuse_b=*/false);
      }
    }
  }

#pragma unroll
  for (int t = 0; t < NT; ++t) {
    const int col = col0 + t * 16 + m;
    const float bv = bias[col];
#pragma unroll
    for (int mt = 0; mt < MT; ++mt) {
#pragma unroll
      for (int r = 0; r < 8; ++r) {
        const int row = row0 + mt * 16 + half * 8 + r;
        C[(size_t)row * N + col] = acc[mt][t][r] + bv;
      }
    }
  }
}

// ---------------------------------------------------------------------------
// Sampling: one wave per (b,q,head); lane == channel within head (Dh == 32).
// softmax over the 4 point-logits is computed redundantly per lane.
// values layout: [B, HW, D] with head channels at h*32+lane -> each gather is
// one contiguous 128B line across the wave.
// ---------------------------------------------------------------------------
__global__ __launch_bounds__(256) void deform_sample_kernel(
    const float* __restrict__ refpts,   // [B,Q,2]
    const float* __restrict__ offsets,  // [B,Q,H,P,2]
    const float* __restrict__ logits,   // [B,Q,H,P]
    const float* __restrict__ values,   // [B,HW,D]
    const int* __restrict__ hptr, const int* __restrict__ wptr,
    float* __restrict__ tmp)            // [B,Q,D]
{
  const int lane = threadIdx.x & 31;
  const int wave = blockIdx.x * 8 + (threadIdx.x >> 5);
  const int head = wave & (N_HEADS - 1);
  const int bq   = wave >> 3;          // b*Q + q
  if (bq >= BSZ * QLEN) return;

  const int Himg = *hptr;
  const int Wimg = *wptr;
  const int HW   = Himg * Wimg;

  const float rx = refpts[(size_t)bq * 2 + 0];
  const float ry = refpts[(size_t)bq * 2 + 1];

  const float* __restrict__ offp = offsets + ((size_t)bq * N_HEADS + head) * (N_POINTS * 2);
  const float* __restrict__ logp = logits  + ((size_t)bq * N_HEADS + head) * N_POINTS;

  const float l0 = logp[0], l1 = logp[1], l2 = logp[2], l3 = logp[3];
  const float mx = fmaxf(fmaxf(l0, l1), fmaxf(l2, l3));
  const float e0 = __expf(l0 - mx), e1 = __expf(l1 - mx);
  const float e2 = __expf(l2 - mx), e3 = __expf(l3 - mx);
  const float inv = 1.0f / (e0 + e1 + e2 + e3);
  const float wgt[N_POINTS] = {e0 * inv, e1 * inv, e2 * inv, e3 * inv};

  const int b = bq / QLEN;
  const float* __restrict__ vbase =
      values + (size_t)b * HW * DMODEL + head * DHEAD + lane;

  float acc = 0.0f;
#pragma unroll
  for (int p = 0; p < N_POINTS; ++p) {
    float lx = rx + offp[p * 2 + 0];
    float ly = ry + offp[p * 2 + 1];
    lx = fminf(fmaxf(lx, 0.0f), 1.0f);
    ly = fminf(fmaxf(ly, 0.0f), 1.0f);
    const float sx = lx * (float)(Wimg - 1);
    const float sy = ly * (float)(Himg - 1);

    int x0 = (int)floorf(sx); x0 = max(0, min(x0, Wimg - 1));
    int y0 = (int)floorf(sy); y0 = max(0, min(y0, Himg - 1));
    const int x1 = min(x0 + 1, Wimg - 1);
    const int y1 = min(y0 + 1, Himg - 1);

    const float wx1 = sx - (float)x0, wx0 = 1.0f - wx1;
    const float wy1 = sy - (float)y0, wy0 = 1.0f - wy1;

    const float g00 = vbase[(size_t)(y0 * Wimg + x0) * DMODEL];
    const float g10 = vbase[(size_t)(y0 * Wimg + x1) * DMODEL];
    const float g01 = vbase[(size_t)(y1 * Wimg + x0) * DMODEL];
    const float g11 = vbase[(size_t)(y1 * Wimg + x1) * DMODEL];

    const float s = g00 * (wx0 * wy0) + g01 * (wx0 * wy1) +
                    g10 * (wx1 * wy0) + g11 * (wx1 * wy1);
    acc = fmaf(s, wgt[p], acc);
  }

  tmp[(size_t)bq * DMODEL + head * DHEAD + lane] = acc;
}

// ---------------------------------------------------------------------------
extern "C" void kernel_launch(void* const* d_in, const int* in_sizes, int n_in,
                              void* d_out, int out_size, void* d_ws, size_t ws_size,
                              hipStream_t stream) {
  const float* query      = (const float*)d_in[0];   // [B,Q,D]
  const float* refpts     = (const float*)d_in[1];   // [B,Q,2]
  const float* input_flat = (const float*)d_in[2];   // [B,HW,D]
  const int*   hptr       = (const int*)d_in[3];
  const int*   wptr       = (const int*)d_in[4];
  const float* W_off      = (const float*)d_in[5];   // [D, H*P*2]
  const float* b_off      = (const float*)d_in[6];
  const float* W_attn     = (const float*)d_in[7];   // [D, H*P]
  const float* b_attn     = (const float*)d_in[8];
  const float* W_val      = (const float*)d_in[9];   // [D, D]
  const float* b_val      = (const float*)d_in[10];
  const float* W_out      = (const float*)d_in[11];  // [D, D]
  const float* b_out      = (const float*)d_in[12];
  float* out = (float*)d_out;

  const int HW = in_sizes[2] / (BSZ * DMODEL);       // 10000

  // workspace partition (floats)
  float* ws      = (float*)d_ws;
  float* values  = ws;                                            // B*HW*D
  float* offsets = values  + (size_t)BSZ * HW * DMODEL;           // B*Q*H*P*2
  float* logits  = offsets + (size_t)BSZ * QLEN * N_HEADS * N_POINTS * 2;
  float* tmp     = logits  + (size_t)BSZ * QLEN * N_HEADS * N_POINTS;  // B*Q*D

  const int Mv = BSZ * HW;     // 80000
  const int Mq = BSZ * QLEN;   // 16384
  const int NOFF = N_HEADS * N_POINTS * 2;  // 64
  const int NATT = N_HEADS * N_POINTS;      // 32
  const int ROWS_PER_BLK = 8 * 16 * 2;      // 8 waves * 32 rows (MT=2) = 256

  // 1) values = input_flatten @ W_val + b_val
  {
    const int rowBlks = (Mv + ROWS_PER_BLK - 1) / ROWS_PER_BLK;  // 313
    const int blocks  = rowBlks * (DMODEL / 64);                 // 1252
    gemm_wmma_f32<2, 4><<<blocks, 256, 0, stream>>>(
        input_flat, W_val, b_val, values, Mv, DMODEL, DMODEL);
  }
  // 2) offsets = query @ W_off + b_off   (N=64 -> one col group)
  {
    const int blocks = (Mq / ROWS_PER_BLK) * (NOFF / 64);        // 64
    gemm_wmma_f32<2, 4><<<blocks, 256, 0, stream>>>(
        query, W_off, b_off, offsets, Mq, NOFF, DMODEL);
  }
  // 3) logits = query @ W_attn + b_attn  (N=32 -> one col group)
  {
    const int blocks = (Mq / ROWS_PER_BLK) * (NATT / 32);        // 64
    gemm_wmma_f32<2, 2><<<blocks, 256, 0, stream>>>(
        query, W_attn, b_attn, logits, Mq, NATT, DMODEL);
  }
  // 4) bilinear sampling + softmax-weighted blend -> tmp [B,Q,D]
  {
    const int waves = Mq * N_HEADS;  // one wave per (b,q,head)
    deform_sample_kernel<<<waves / 8, 256, 0, stream>>>(refpts, offsets, logits,
                                                        values, hptr, wptr, tmp);
  }
  // 5) out = tmp @ W_out + b_out
  {
    const int blocks = (Mq / ROWS_PER_BLK) * (DMODEL / 64);      // 256
    gemm_wmma_f32<2, 4><<<blocks, 256, 0, stream>>>(
        tmp, W_out, b_out, out, Mq, DMODEL, DMODEL);
  }
}